// LTCCell_64536178590132
// MI455X (gfx1250) — compile-verified
//
#include <hip/hip_runtime.h>
#include <hip/hip_bf16.h>

#define LTC_IN   256
#define LTC_U    512
#define UNFOLDS  6
#define ROWS     4          // batch rows per workgroup

typedef unsigned int u32x4 __attribute__((ext_vector_type(4)));
typedef int          i32x8 __attribute__((ext_vector_type(8)));
typedef int          i32x4 __attribute__((ext_vector_type(4)));

// fast sigmoid: 1/(1+exp2(-t*log2(e))) -> v_exp_f32 + v_rcp_f32 (trans pipe)
__device__ __forceinline__ float fsig(float t) {
    float e = __builtin_amdgcn_exp2f(-1.44269504088896341f * t);
    return __builtin_amdgcn_rcpf(1.0f + e);
}

#if defined(__has_builtin)
#if __has_builtin(__builtin_amdgcn_tensor_load_to_lds)
#define HAVE_TDM 1
#endif
#endif

#ifdef HAVE_TDM
// 2D tile load via Tensor Data Mover: rows x width f32 elements, row stride
// `stride_elts`, global -> LDS (row-major, densely packed in LDS).
// D# layout per CDNA5 ISA.
// This toolchain's builtin takes 6 args (g0, g1, g2, g3, g4, cpol).
__device__ __forceinline__ void tdm_load_2d(unsigned lds_addr, const void* gptr,
                                            unsigned width_elts, unsigned rows,
                                            unsigned stride_elts) {
    unsigned long long ga = (unsigned long long)(size_t)gptr;
    u32x4 g0;
    g0[0] = 1u;                                   // count=1 (valid user descriptor)
    g0[1] = lds_addr;                             // lds_addr [63:32]
    g0[2] = (unsigned)(ga & 0xFFFFFFFFu);         // global_addr[31:0]
    g0[3] = (unsigned)((ga >> 32) & 0x01FFFFFFu)  // global_addr[56:32]
          | (2u << 30);                           // type = 2 ("image")
    i32x8 g1;
    g1[0] = (int)(2u << 16);                      // data_size=2 -> 4 bytes; wg_mask=0
    g1[1] = (int)((width_elts & 0xFFFFu) << 16);  // tensor_dim0[15:0]
    g1[2] = (int)((width_elts >> 16) | ((rows & 0xFFFFu) << 16)); // dim0[31:16], dim1[15:0]
    g1[3] = (int)((rows >> 16) | ((width_elts & 0xFFFFu) << 16)); // dim1[31:16], tile_dim0
    g1[4] = (int)(rows & 0xFFFFu);                // tile_dim1 (tile_dim2 = 0)
    g1[5] = (int)stride_elts;                     // tensor_dim0_stride[31:0]
    g1[6] = 0;                                    // stride[47:32]=0, dim1_stride lo=0
    g1[7] = 0;
    i32x4 z4 = {0, 0, 0, 0};                      // groups 2/3 unused (2D tensor)
    i32x8 z8 = {0, 0, 0, 0, 0, 0, 0, 0};          // trailing group (this toolchain)
    __builtin_amdgcn_tensor_load_to_lds(g0, g1, z4, z4, z8, 0);
}
#endif

__global__ void __launch_bounds__(LTC_U)
ltc_cell_kernel(const float* __restrict__ inputs,   // [B, IN]
                const float* __restrict__ state,    // [B, U]
                const float* __restrict__ input_w,  // [IN]
                const float* __restrict__ input_b,  // [IN]
                const float* __restrict__ smu,      // [IN, U]
                const float* __restrict__ ssig,     // [IN, U]
                const float* __restrict__ sW,       // [IN, U]
                const float* __restrict__ serev,    // [IN, U]
                const float* __restrict__ mu,       // [U, U]
                const float* __restrict__ sigma,    // [U, U]
                const float* __restrict__ Wm,       // [U, U]
                const float* __restrict__ erev,     // [U, U]
                const float* __restrict__ vleak,    // [U]
                const float* __restrict__ gleak,    // [U]
                const float* __restrict__ cm_t,     // [U]
                float* __restrict__ out)            // [B, U]
{
    __shared__ __align__(16) float v_in[ROWS * LTC_U];   // TDM staging: [r][j]
    __shared__ __align__(16) float x_in[ROWS * LTC_IN];  // TDM staging: [r][i]
    __shared__ __align__(16) float x_s[LTC_IN * ROWS];   // transposed:  [i][r]
    __shared__ __align__(16) float v_s[2][LTC_U * ROWS]; // transposed, double-buffered

    const int tid = threadIdx.x;          // == u
    const int u   = tid;
    const int b0  = blockIdx.x * ROWS;

    // ---- stage inputs & state tiles into LDS (TDM on gfx1250) ----
#ifdef HAVE_TDM
    if (__builtin_amdgcn_readfirstlane((int)(threadIdx.x >> 5)) == 0) {
        tdm_load_2d((unsigned)(size_t)(void*)v_in, state  + (size_t)b0 * LTC_U,
                    LTC_U,  ROWS, LTC_U);
        tdm_load_2d((unsigned)(size_t)(void*)x_in, inputs + (size_t)b0 * LTC_IN,
                    LTC_IN, ROWS, LTC_IN);
        __builtin_amdgcn_s_wait_tensorcnt(0);
    }
#else
    for (int k = tid; k < ROWS * LTC_U;  k += blockDim.x) v_in[k] = state [(size_t)b0 * LTC_U  + k];
    for (int k = tid; k < ROWS * LTC_IN; k += blockDim.x) x_in[k] = inputs[(size_t)b0 * LTC_IN + k];
#endif
    __syncthreads();

    // ---- transpose to [index][row] so the hot loop does one b128 LDS broadcast ----
    for (int k = tid; k < LTC_U * ROWS; k += blockDim.x) {
        int j = k >> 2, r = k & 3;
        v_s[0][k] = v_in[r * LTC_U + j];
    }
    for (int k = tid; k < LTC_IN * ROWS; k += blockDim.x) {
        int i = k >> 2, r = k & 3;
        x_s[k] = fmaf(x_in[r * LTC_IN + i], input_w[i], input_b[i]);
    }
    __syncthreads();

    // ---- sensory synapses (once) ----
    float nums[ROWS] = {0.f, 0.f, 0.f, 0.f};
    float dens[ROWS] = {0.f, 0.f, 0.f, 0.f};
    {
        const float* wp = sW    + u;
        const float* mp = smu   + u;
        const float* sp = ssig  + u;
        const float* ep = serev + u;
        #pragma unroll 4
        for (int i = 0; i < LTC_IN; ++i) {
            float w  = wp[(size_t)i * LTC_U];
            float m  = mp[(size_t)i * LTC_U];
            float sg = sp[(size_t)i * LTC_U];
            float er = ep[(size_t)i * LTC_U];
            float4 xv = *reinterpret_cast<const float4*>(x_s + (i << 2));
            float xr[ROWS] = {xv.x, xv.y, xv.z, xv.w};
            #pragma unroll
            for (int r = 0; r < ROWS; ++r) {
                float a = w * fsig(sg * (xr[r] - m));
                nums[r] = fmaf(a, er, nums[r]);
                dens[r] += a;
            }
        }
    }

    // per-unit constants
    const float cm   = cm_t[u];
    const float gl   = gleak[u];
    const float glvl = gl * vleak[u];

    float vown[ROWS];
    #pragma unroll
    for (int r = 0; r < ROWS; ++r) vown[r] = v_s[0][(u << 2) + r];

    // ---- 6 semi-implicit unfolds ----
    int cur = 0;
    for (int step = 0; step < UNFOLDS; ++step) {
        float num[ROWS], den[ROWS];
        #pragma unroll
        for (int r = 0; r < ROWS; ++r) { num[r] = nums[r]; den[r] = dens[r]; }

        const float* vbuf = v_s[cur];
        const float* wp = Wm    + u;
        const float* mp = mu    + u;
        const float* sp = sigma + u;
        const float* ep = erev  + u;
        #pragma unroll 4
        for (int j = 0; j < LTC_U; ++j) {
            float w  = wp[(size_t)j * LTC_U];
            float m  = mp[(size_t)j * LTC_U];
            float sg = sp[(size_t)j * LTC_U];
            float er = ep[(size_t)j * LTC_U];
            float4 vv = *reinterpret_cast<const float4*>(vbuf + (j << 2));
            float vj[ROWS] = {vv.x, vv.y, vv.z, vv.w};
            #pragma unroll
            for (int r = 0; r < ROWS; ++r) {
                float a = w * fsig(sg * (vj[r] - m));
                num[r] = fmaf(a, er, num[r]);
                den[r] += a;
            }
        }

        float* vnext = v_s[cur ^ 1];
        #pragma unroll
        for (int r = 0; r < ROWS; ++r) {
            float vn = (fmaf(cm, vown[r], glvl) + num[r]) / (cm + gl + den[r]);
            vown[r] = vn;
            vnext[(u << 2) + r] = vn;
        }
        __syncthreads();          // gfx1250 split barrier
        cur ^= 1;
    }

    // ---- write result [B, U], coalesced across u ----
    #pragma unroll
    for (int r = 0; r < ROWS; ++r)
        out[(size_t)(b0 + r) * LTC_U + u] = vown[r];
}

extern "C" void kernel_launch(void* const* d_in, const int* in_sizes, int n_in,
                              void* d_out, int out_size, void* d_ws, size_t ws_size,
                              hipStream_t stream) {
    const float* inputs   = (const float*)d_in[0];
    const float* state    = (const float*)d_in[1];
    const float* input_w  = (const float*)d_in[2];
    const float* input_b  = (const float*)d_in[3];
    const float* smu      = (const float*)d_in[4];
    const float* ssig     = (const float*)d_in[5];
    const float* sW       = (const float*)d_in[6];
    const float* serev    = (const float*)d_in[7];
    const float* mu       = (const float*)d_in[8];
    const float* sigma    = (const float*)d_in[9];
    const float* Wm       = (const float*)d_in[10];
    const float* erev     = (const float*)d_in[11];
    const float* vleak    = (const float*)d_in[12];
    const float* gleak    = (const float*)d_in[13];
    const float* cm_t     = (const float*)d_in[14];
    float* out            = (float*)d_out;

    const int B = in_sizes[0] / LTC_IN;       // 512
    dim3 grid(B / ROWS);                      // 128 workgroups
    dim3 block(LTC_U);                        // 512 threads = 16 wave32

    ltc_cell_kernel<<<grid, block, 0, stream>>>(
        inputs, state, input_w, input_b, smu, ssig, sW, serev,
        mu, sigma, Wm, erev, vleak, gleak, cm_t, out);
}